// HSS_76398878261835
// MI455X (gfx1250) — compile-verified
//
#include <hip/hip_runtime.h>
#include <hip/hip_bf16.h>
#include <math.h>

// Problem constants (from reference)
#define BB   32
#define CC   256
#define HH   64
#define WW_  64
#define MIP  8
#define OUP  256
#define SS   128            // H + W
#define BN_EPS 1e-5f

typedef float v2f __attribute__((ext_vector_type(2)));
typedef float v8f __attribute__((ext_vector_type(8)));

// ---------------------------------------------------------------------------
// Kernel 1: per (b,c) slice reductions.
// One 256-thread block per 64x64 slice (8192 blocks). Slice staged in padded
// LDS (stride 65 -> bank-conflict-free row and column walks).
// Emits yin[b][c][s], s in [0,128): s<64 -> a_h[h], s>=64 -> a_w[w].
//   a_h[h] = rowmean[h] + softmax_h(rowmax)[h] / 64
//   a_w[w] = colmean[w] + softmax_w(colmax)[w] / 64
// ---------------------------------------------------------------------------
__global__ void k1_reduce(const float* __restrict__ x, float* __restrict__ yin) {
    __shared__ float tile[HH * 65];
    __shared__ float rmax[HH], rsum[HH], cmax[WW_], csum[WW_];
    __shared__ float gmax[2];   // [0]=max over rowmax, [1]=max over colmax
    __shared__ float gsum[2];   // softmax partition sums

    const int slice = blockIdx.x;                 // b*C + c
    const float* xs = x + (size_t)slice * (HH * WW_);
    const int t = threadIdx.x;

    // Coalesced load of the 16KB slice into LDS with padding.
    for (int i = t; i < HH * WW_; i += 256) {
        int h = i >> 6, w = i & 63;
        tile[h * 65 + w] = xs[i];
    }
    __syncthreads();

    if (t < 64) {                                 // row stats (row = t)
        float m = -3.402823466e38f, s = 0.f;
        #pragma unroll 8
        for (int w = 0; w < WW_; ++w) {
            float v = tile[t * 65 + w];
            m = fmaxf(m, v); s += v;
        }
        rmax[t] = m; rsum[t] = s;
    } else if (t < 128) {                         // col stats (col = t-64)
        int w = t - 64;
        float m = -3.402823466e38f, s = 0.f;
        #pragma unroll 8
        for (int h = 0; h < HH; ++h) {
            float v = tile[h * 65 + w];
            m = fmaxf(m, v); s += v;
        }
        cmax[w] = m; csum[w] = s;
    }
    __syncthreads();

    if (t == 0) { float m = -3.402823466e38f; for (int i = 0; i < HH;  ++i) m = fmaxf(m, rmax[i]); gmax[0] = m; }
    if (t == 1) { float m = -3.402823466e38f; for (int i = 0; i < WW_; ++i) m = fmaxf(m, cmax[i]); gmax[1] = m; }
    __syncthreads();

    if (t < 64)        rmax[t]      = expf(rmax[t]      - gmax[0]);
    else if (t < 128)  cmax[t - 64] = expf(cmax[t - 64] - gmax[1]);
    __syncthreads();

    if (t == 0) { float s = 0.f; for (int i = 0; i < HH;  ++i) s += rmax[i]; gsum[0] = s; }
    if (t == 1) { float s = 0.f; for (int i = 0; i < WW_; ++i) s += cmax[i]; gsum[1] = s; }
    __syncthreads();

    const float inv64 = 1.0f / 64.0f;
    if (t < 64) {
        yin[(size_t)slice * SS + t] = rsum[t] * inv64 + (rmax[t] / gsum[0]) * inv64;
    } else if (t < 128) {
        int w = t - 64;
        yin[(size_t)slice * SS + 64 + w] = csum[w] * inv64 + (cmax[w] / gsum[1]) * inv64;
    }
}

// ---------------------------------------------------------------------------
// Kernel 2: per batch (32 blocks x 256 threads = 8 waves):
//   GEMM1 (8x256 @ 256x128) on VALU -> BN + hswish -> y2[8][128] in LDS,
//   then GEMM2 pair ((256x8)@(8x64) with wh / ww) via V_WMMA_F32_16X16X4_F32,
//   sigmoid, scatter gates g[b][o][s].
// ---------------------------------------------------------------------------
__global__ void k2_gemms(const float* __restrict__ yin,
                         const float* __restrict__ w1,
                         const float* __restrict__ bn_gamma,
                         const float* __restrict__ bn_beta,
                         const float* __restrict__ bn_mean,
                         const float* __restrict__ bn_var,
                         const float* __restrict__ wh,
                         const float* __restrict__ ww,
                         float* __restrict__ g) {
    __shared__ float w1s[MIP * CC];     // 8KB
    __shared__ float whs[OUP * MIP];    // 8KB
    __shared__ float wws[OUP * MIP];    // 8KB
    __shared__ float y2 [MIP * SS];     // 4KB

    const int b = blockIdx.x;
    const int t = threadIdx.x;

    for (int i = t; i < MIP * CC; i += 256) {
        w1s[i] = w1[i];
        whs[i] = wh[i];
        wws[i] = ww[i];
    }
    __syncthreads();

    // --- GEMM1 + BN + hswish: threads 0..127 each own a column s ---
    if (t < SS) {
        float acc[MIP];
        #pragma unroll
        for (int m = 0; m < MIP; ++m) acc[m] = 0.f;
        const float* yb = yin + (size_t)b * CC * SS + t;
        for (int c = 0; c < CC; ++c) {
            float v = yb[(size_t)c * SS];
            #pragma unroll
            for (int m = 0; m < MIP; ++m) acc[m] = fmaf(w1s[m * CC + c], v, acc[m]);
        }
        #pragma unroll
        for (int m = 0; m < MIP; ++m) {
            float sc = bn_gamma[m] * rsqrtf(bn_var[m] + BN_EPS);
            float sh = bn_beta[m] - bn_mean[m] * sc;
            float yv = fmaf(acc[m], sc, sh);
            float hs = yv * fminf(fmaxf(yv + 3.f, 0.f), 6.f) * (1.f / 6.f);
            y2[m * SS + t] = hs;
        }
    }
    __syncthreads();

    // --- GEMM2 via WMMA f32 16x16x4 (K=8 -> two k-steps), full EXEC ---
    const int wv   = t >> 5;        // wave id (8 waves)
    const int lane = t & 31;
    const int half = lane >> 4;     // lane-half selects K group per ISA layout
    const int lm   = lane & 15;
    float* gb = g + (size_t)b * OUP * SS;

    // 128 tiles: {gate(2)} x {o_tile(16)} x {n_tile(4)}
    for (int tileid = wv; tileid < 128; tileid += 8) {
        const int gsel  = tileid >> 6;
        const int rem   = tileid & 63;
        const int ot    = rem >> 2;
        const int nt    = rem & 3;
        const float* W2 = gsel ? wws : whs;
        const int sbase = gsel ? 64 : 0;

        const int row = ot * 16 + lm;          // A: M = row, K per lane-half
        v2f a0, a1, b0, b1;
        a0.x = W2[row * MIP + 2 * half + 0];   // K = 0/2
        a0.y = W2[row * MIP + 2 * half + 1];   // K = 1/3
        a1.x = W2[row * MIP + 2 * half + 4];   // K = 4/6
        a1.y = W2[row * MIP + 2 * half + 5];   // K = 5/7

        const int s = sbase + nt * 16 + lm;    // B: N = lm, K per lane-half
        b0.x = y2[(2 * half + 0) * SS + s];
        b0.y = y2[(2 * half + 1) * SS + s];
        b1.x = y2[(2 * half + 4) * SS + s];
        b1.y = y2[(2 * half + 5) * SS + s];

        v8f c = {};
        c = __builtin_amdgcn_wmma_f32_16x16x4_f32(false, a0, false, b0,
                                                  (short)0, c, false, false);
        c = __builtin_amdgcn_wmma_f32_16x16x4_f32(false, a1, false, b1,
                                                  (short)0, c, false, false);

        // D layout: VGPR r -> M = r (lanes 0-15) / 8+r (lanes 16-31), N = lm
        #pragma unroll
        for (int r = 0; r < 8; ++r) {
            int orow = ot * 16 + r + half * 8;
            float v = c[r];
            gb[(size_t)orow * SS + s] = 1.f / (1.f + expf(-v));
        }
    }
}

// ---------------------------------------------------------------------------
// Kernel 3: out[b,c,h,w] = x[b,c,h,w] * g[b,c,h] * g[b,c,64+w]
// float4 per thread, fully coalesced. x expected L2-resident (134MB < 192MB).
// ---------------------------------------------------------------------------
__global__ void k3_apply(const float* __restrict__ x,
                         const float* __restrict__ g,
                         float* __restrict__ out) {
    const size_t i = ((size_t)blockIdx.x * blockDim.x + threadIdx.x) * 4;
    const int w   = (int)(i & 63);
    const int bch = (int)(i >> 6);
    const int h   = bch & 63;
    const int bc  = bch >> 6;

    float4 xv = *(const float4*)(x + i);
    const float gh = g[(size_t)bc * SS + h];
    const float* gw = g + (size_t)bc * SS + 64 + w;

    float4 o;
    o.x = xv.x * gh * gw[0];
    o.y = xv.y * gh * gw[1];
    o.z = xv.z * gh * gw[2];
    o.w = xv.w * gh * gw[3];
    *(float4*)(out + i) = o;
}

// ---------------------------------------------------------------------------
extern "C" void kernel_launch(void* const* d_in, const int* in_sizes, int n_in,
                              void* d_out, int out_size, void* d_ws, size_t ws_size,
                              hipStream_t stream) {
    const float* x        = (const float*)d_in[0];
    const float* w1       = (const float*)d_in[1];
    const float* bn_gamma = (const float*)d_in[2];
    const float* bn_beta  = (const float*)d_in[3];
    const float* bn_mean  = (const float*)d_in[4];
    const float* bn_var   = (const float*)d_in[5];
    const float* wh       = (const float*)d_in[6];
    const float* ww       = (const float*)d_in[7];
    float* out = (float*)d_out;

    // Workspace carve-out: yin = B*C*128 floats (4MB), g = B*OUP*128 floats (4MB)
    float* yin = (float*)d_ws;
    float* gg  = yin + (size_t)BB * CC * SS;

    k1_reduce<<<BB * CC, 256, 0, stream>>>(x, yin);
    k2_gemms <<<BB, 256, 0, stream>>>(yin, w1, bn_gamma, bn_beta, bn_mean, bn_var,
                                      wh, ww, gg);
    const size_t total = (size_t)BB * CC * HH * WW_;
    k3_apply <<<(unsigned)(total / 4 / 256), 256, 0, stream>>>(x, gg, out);
}